// bigbrain_15040975471265
// MI455X (gfx1250) — compile-verified
//
#include <hip/hip_runtime.h>
#include <math.h>

#define BATCH   8192
#define DIM     16
#define DIM1    48
#define TTH     20
#define VOCAB   32000
#define NTILES  (VOCAB / 16)   // 2000 column tiles
#define NRB     (BATCH / 16)   // 512 row blocks

typedef __attribute__((ext_vector_type(2))) float v2f;
typedef __attribute__((ext_vector_type(8))) float v8f;

// ---------------------------------------------------------------------------
// Phase 1: per-row tiny MLP.  One thread handles one batch row end-to-end and
// writes o[row][0..15] to the workspace.  ~14K FMAs/thread, 8192 threads:
// negligible next to the 45us store floor of phase 2.
// ---------------------------------------------------------------------------
__device__ __forceinline__ void ln_silu(float* h, const float* __restrict__ g,
                                        const float* __restrict__ b) {
  float mu = 0.f;
  for (int i = 0; i < DIM1; ++i) mu += h[i];
  mu *= (1.0f / DIM1);
  float var = 0.f;
  for (int i = 0; i < DIM1; ++i) { float d = h[i] - mu; var += d * d; }
  var *= (1.0f / DIM1);
  float r = rsqrtf(var + 1e-5f);
  for (int i = 0; i < DIM1; ++i) {
    float y = (h[i] - mu) * r * g[i] + b[i];
    h[i] = y / (1.f + expf(-y));           // silu
  }
}

__global__ void __launch_bounds__(256) bigbrain_phase1(
    const float* __restrict__ we,  const float* __restrict__ cx1,
    const float* __restrict__ th,
    const float* __restrict__ WI1, const float* __restrict__ bI1,
    const float* __restrict__ WA1, const float* __restrict__ bA1,
    const float* __restrict__ WM1, const float* __restrict__ bM1,
    const float* __restrict__ WM2, const float* __restrict__ bM2,
    const float* __restrict__ WM3, const float* __restrict__ bM3,
    const float* __restrict__ lng, const float* __restrict__ lnb,
    const float* __restrict__ WO1, const float* __restrict__ bO1,
    float* __restrict__ o_out)
{
  const int row = blockIdx.x * blockDim.x + threadIdx.x;
  if (row >= BATCH) return;

  float x[DIM], cx[DIM];
#pragma unroll
  for (int i = 0; i < DIM; ++i) { x[i] = we[row * DIM + i]; cx[i] = cx1[row * DIM + i]; }

  // out = I1(word_embed)                                  [48]
  float out48[DIM1];
  for (int j = 0; j < DIM1; ++j) {
    float s = bI1[j];
    const float* w = WI1 + j * DIM;
#pragma unroll
    for (int k = 0; k < DIM; ++k) s += w[k] * x[k];
    out48[j] = s;
  }

  // out_proj = A1(out)                                    [16]
  float op[DIM];
  for (int j = 0; j < DIM; ++j) {
    float s = bA1[j];
    const float* w = WA1 + j * DIM1;
    for (int k = 0; k < DIM1; ++k) s += w[k] * out48[k];
    op[j] = s;
  }

  // h = silu(LN(M1(cat[out, context1, out_proj])))        [48]
  float h[DIM1];
  for (int j = 0; j < DIM1; ++j) {
    const float* w = WM1 + j * 80;
    float s = bM1[j];
    for (int k = 0; k < DIM1; ++k) s += w[k] * out48[k];
#pragma unroll
    for (int k = 0; k < DIM; ++k) s += w[48 + k] * cx[k];
#pragma unroll
    for (int k = 0; k < DIM; ++k) s += w[64 + k] * op[k];
    h[j] = s;
  }
  ln_silu(h, lng, lnb);

  // scores = thoughts1 . out_proj ; top-3 (ties -> lowest index, like top_k)
  float sc[TTH];
  const float* tb = th + (size_t)row * TTH * DIM;
  for (int t = 0; t < TTH; ++t) {
    float s = 0.f;
#pragma unroll
    for (int k = 0; k < DIM; ++k) s += tb[t * DIM + k] * op[k];
    sc[t] = s;
  }
  float tf[3 * DIM];
  for (int sel = 0; sel < 3; ++sel) {
    float best = -INFINITY; int bi = 0;
    for (int t = 0; t < TTH; ++t)
      if (sc[t] > best) { best = sc[t]; bi = t; }
    sc[bi] = -INFINITY;
#pragma unroll
    for (int k = 0; k < DIM; ++k) tf[sel * DIM + k] = tb[bi * DIM + k];
  }

  // h2 = silu(LN(M2(cat[h, top_flat, context1])))         [48]
  float h2[DIM1];
  for (int j = 0; j < DIM1; ++j) {
    const float* w = WM2 + j * 112;
    float s = bM2[j];
    for (int k = 0; k < DIM1; ++k) s += w[k] * h[k];
    for (int k = 0; k < 3 * DIM; ++k) s += w[48 + k] * tf[k];
#pragma unroll
    for (int k = 0; k < DIM; ++k) s += w[96 + k] * cx[k];
    h2[j] = s;
  }
  ln_silu(h2, lng, lnb);

  // h3 = silu(LN(M3(h2)))                                 [48]
  for (int j = 0; j < DIM1; ++j) {
    const float* w = WM3 + j * DIM1;
    float s = bM3[j];
    for (int k = 0; k < DIM1; ++k) s += w[k] * h2[k];
    h[j] = s;
  }
  ln_silu(h, lng, lnb);

  // o = O1(h3)                                            [16]
  for (int j = 0; j < DIM; ++j) {
    const float* w = WO1 + j * DIM1;
    float s = bO1[j];
    for (int k = 0; k < DIM1; ++k) s += w[k] * h[k];
    o_out[row * DIM + j] = s;
  }
}

// ---------------------------------------------------------------------------
// Phase 2: logits = o[8192,16] @ embed_w.T[16,32000] via f32 WMMA.
// Grid: 512 row-blocks x 8 waves; each wave holds one 16x16 A tile and sweeps
// 250 column tiles (16 cols of embed_w each), 4 chained v_wmma_f32_16x16x4_f32
// per tile (K = 16).  Output streamed with non-temporal stores (~1.05 GB ->
// ~45 us at 23.3 TB/s; embed_w stays hot in L2).
//
// f32 A(16x4) fragment, lane L: M = L&15, K-pair = {4s, 4s+1} for L<16,
// {4s+2, 4s+3} for L>=16.  B(4x16) mirrors with lane = N.  C/D: lane L,
// vgpr v -> M = v + 8*(L>=16), N = L&15.
// ---------------------------------------------------------------------------
__global__ void __launch_bounds__(256) bigbrain_phase2(
    const float* __restrict__ o, const float* __restrict__ E,
    float* __restrict__ out)
{
  const int lane = threadIdx.x & 31;
  const int wave = threadIdx.x >> 5;       // 0..7, uniform within a wave
  const int rb   = blockIdx.x;             // 0..511
  const int m    = lane & 15;
  const int koff = (lane < 16) ? 0 : 2;

  // A fragments for the 4 K-steps: o row (rb*16 + m), K offsets {4s,4s+1}+koff
  const float* arow = o + (size_t)(rb * 16 + m) * DIM + koff;
  v2f a0 = *(const v2f*)(arow + 0);
  v2f a1 = *(const v2f*)(arow + 4);
  v2f a2 = *(const v2f*)(arow + 8);
  v2f a3 = *(const v2f*)(arow + 12);

  const int mbase = (lane < 16) ? 0 : 8;

  for (int t = wave; t < NTILES; t += 8) {
    // B fragments: embed_w row (t*16 + m) supplies column N = m of B tile
    const float* brow = E + (size_t)(t * 16 + m) * DIM + koff;
    v2f b0 = *(const v2f*)(brow + 0);
    v2f b1 = *(const v2f*)(brow + 4);
    v2f b2 = *(const v2f*)(brow + 8);
    v2f b3 = *(const v2f*)(brow + 12);

    v8f c = {};
    c = __builtin_amdgcn_wmma_f32_16x16x4_f32(false, a0, false, b0, (short)0, c, false, false);
    c = __builtin_amdgcn_wmma_f32_16x16x4_f32(false, a1, false, b1, (short)0, c, false, false);
    c = __builtin_amdgcn_wmma_f32_16x16x4_f32(false, a2, false, b2, (short)0, c, false, false);
    c = __builtin_amdgcn_wmma_f32_16x16x4_f32(false, a3, false, b3, (short)0, c, false, false);

    // scatter D: logits[rb*16 + mbase + v][t*16 + m]
    float* obase = out + (size_t)(rb * 16 + mbase) * VOCAB + t * 16 + m;
#pragma unroll
    for (int v = 0; v < 8; ++v)
      __builtin_nontemporal_store(c[v], obase + (size_t)v * VOCAB);
  }
}

extern "C" void kernel_launch(void* const* d_in, const int* in_sizes, int n_in,
                              void* d_out, int out_size, void* d_ws, size_t ws_size,
                              hipStream_t stream) {
  const float* we   = (const float*)d_in[0];
  const float* cx1  = (const float*)d_in[1];
  const float* th   = (const float*)d_in[2];
  const float* WI1  = (const float*)d_in[3];
  const float* bI1  = (const float*)d_in[4];
  const float* WA1  = (const float*)d_in[5];
  const float* bA1  = (const float*)d_in[6];
  const float* WM1  = (const float*)d_in[7];
  const float* bM1  = (const float*)d_in[8];
  const float* WM2  = (const float*)d_in[9];
  const float* bM2  = (const float*)d_in[10];
  const float* WM3  = (const float*)d_in[11];
  const float* bM3  = (const float*)d_in[12];
  const float* lng  = (const float*)d_in[13];
  const float* lnb  = (const float*)d_in[14];
  const float* WO1  = (const float*)d_in[15];
  const float* bO1  = (const float*)d_in[16];
  const float* Ew   = (const float*)d_in[17];

  float* o_buf = (float*)d_ws;             // BATCH*16 floats = 512 KB

  bigbrain_phase1<<<BATCH / 256, 256, 0, stream>>>(
      we, cx1, th, WI1, bI1, WA1, bA1, WM1, bM1, WM2, bM2, WM3, bM3,
      lng, lnb, WO1, bO1, o_buf);

  bigbrain_phase2<<<NRB, 256, 0, stream>>>(o_buf, Ew, (float*)d_out);
}